// NerfRender_89790586290646
// MI455X (gfx1250) — compile-verified
//
#include <hip/hip_runtime.h>
#include <hip/hip_bf16.h>

#define NSAMP 192
#define NT    193
#define SORTN 256

// ---- CDNA5 async global->LDS copy support (guarded; fallback = plain loads) ----
#if defined(__has_builtin)
#  if __has_builtin(__builtin_amdgcn_global_load_async_to_lds_b128) && \
      __has_builtin(__builtin_amdgcn_global_load_async_to_lds_b32)
#    define HAVE_ASYNC_LDS 1
#  endif
#  if __has_builtin(__builtin_amdgcn_s_wait_asynccnt)
#    define HAVE_WAIT_ASYNC_BUILTIN 1
#  endif
#endif

typedef int v4i __attribute__((ext_vector_type(4)));

typedef __attribute__((address_space(1))) int  gi32_t;   // global (AS1) int
typedef __attribute__((address_space(3))) int  li32_t;   // LDS (AS3) int
typedef __attribute__((address_space(1))) v4i  gv4i_t;   // global (AS1) int4
typedef __attribute__((address_space(3))) v4i  lv4i_t;   // LDS (AS3) int4

// Flat->AS1 is value-preserving; flat LDS address keeps the LDS byte offset in
// the low 32 bits (aperture occupies the upper dword), so truncation gives AS3.
#define GI32(p)  ((gi32_t*)(unsigned long long)(size_t)(p))
#define LI32(p)  ((li32_t*)(unsigned)(size_t)(p))
#define GI128(p) ((gv4i_t*)(unsigned long long)(size_t)(p))
#define LI128(p) ((lv4i_t*)(unsigned)(size_t)(p))

__device__ __forceinline__ void wait_async0() {
#if defined(HAVE_WAIT_ASYNC_BUILTIN)
    __builtin_amdgcn_s_wait_asynccnt(0);
#else
    asm volatile("s_wait_asynccnt 0" ::: "memory");
#endif
}

// One wave32 per ray; single-wave workgroup (barriers degrade to S_NOP + dscnt wait).
__global__ __launch_bounds__(32) void nerf_integrate_kernel(
    const float* __restrict__ t_g,
    const float* __restrict__ sig_g,
    const float* __restrict__ c_g,
    float* __restrict__ rgb_out,
    float* __restrict__ wi_out)
{
    __shared__ __align__(16) float s_t[SORTN];      // 193 real + inf pad
    __shared__ __align__(16) float s_sig[NSAMP];
    __shared__ __align__(16) float s_c[NSAMP * 3];

    const int ray  = blockIdx.x;
    const int lane = threadIdx.x;

    const float* tg = t_g  + (size_t)ray * NT;          // 4B-aligned rows only
    const float* sg = sig_g + (size_t)ray * NSAMP;      // 16B-aligned rows
    const float* cg = c_g  + (size_t)ray * (NSAMP * 3); // 16B-aligned rows

#if defined(HAVE_ASYNC_LDS)
    // t: 193 dwords, rows only dword-aligned -> 7 predicated async b32 rounds.
    #pragma unroll
    for (int r = 0; r < 7; ++r) {
        int i = r * 32 + lane;
        if (i < NT)
            __builtin_amdgcn_global_load_async_to_lds_b32(GI32(tg + i), LI32(&s_t[i]), 0, 0);
    }
    // sigma: 192 dwords = 1.5 * (32 lanes * 16B) -> b128 all lanes + b128 lanes<16.
    __builtin_amdgcn_global_load_async_to_lds_b128(GI128(sg + lane * 4),
                                                   LI128(&s_sig[lane * 4]), 0, 0);
    if (lane < 16)
        __builtin_amdgcn_global_load_async_to_lds_b128(GI128(sg + 128 + lane * 4),
                                                       LI128(&s_sig[128 + lane * 4]), 0, 0);
    // c: 576 dwords = 4.5 * 512B -> 4 full b128 waves + b128 lanes<16.
    #pragma unroll
    for (int k = 0; k < 4; ++k)
        __builtin_amdgcn_global_load_async_to_lds_b128(GI128(cg + k * 128 + lane * 4),
                                                       LI128(&s_c[k * 128 + lane * 4]), 0, 0);
    if (lane < 16)
        __builtin_amdgcn_global_load_async_to_lds_b128(GI128(cg + 512 + lane * 4),
                                                       LI128(&s_c[512 + lane * 4]), 0, 0);
#else
    for (int i = lane; i < NT; i += 32)        s_t[i]   = tg[i];
    for (int i = lane; i < NSAMP; i += 32)     s_sig[i] = sg[i];
    for (int i = lane; i < NSAMP * 3; i += 32) s_c[i]   = cg[i];
#endif

    // Pad sort buffer with +inf (disjoint from async destinations).
    for (int i = NT + lane; i < SORTN; i += 32)
        s_t[i] = __builtin_inff();

#if defined(HAVE_ASYNC_LDS)
    wait_async0();
#endif
    __syncthreads();

    // ---- Bitonic sort of 256 floats in LDS (ascending). Pairs within a pass are
    // disjoint, so only inter-pass barriers are required. ----
    for (int k = 2; k <= SORTN; k <<= 1) {
        for (int j = k >> 1; j > 0; j >>= 1) {
            #pragma unroll
            for (int w = 0; w < SORTN / 32; ++w) {
                int i = w * 32 + lane;
                int p = i ^ j;
                if (p > i) {
                    float a = s_t[i];
                    float b = s_t[p];
                    bool descend = (i & k) != 0;
                    bool swap = descend ? (a < b) : (a > b);
                    if (swap) { s_t[i] = b; s_t[p] = a; }
                }
            }
            __syncthreads();
        }
    }

    // ---- Integration: lane owns 6 contiguous samples [6*lane, 6*lane+6). ----
    const int i0 = lane * 6;
    float sdt[6];
    float lsum = 0.f;
    #pragma unroll
    for (int q = 0; q < 6; ++q) {
        int i = i0 + q;
        float dt = s_t[i + 1] - s_t[i];
        sdt[q] = s_sig[i] * dt;
        lsum += sdt[q];
    }

    // Wave32 inclusive scan of per-lane sums -> exclusive prefix for this lane.
    float incl = lsum;
    #pragma unroll
    for (int off = 1; off < 32; off <<= 1) {
        float u = __shfl_up(incl, (unsigned)off, 32);
        if (lane >= off) incl += u;
    }
    float cum = incl - lsum;   // sum of all sdt before this lane's first sample

    float r0 = 0.f, r1 = 0.f, r2 = 0.f;
    float w[6];
    #pragma unroll
    for (int q = 0; q < 6; ++q) {
        int i = i0 + q;
        float T     = expf(-cum);           // transmittance before sample i
        float alpha = 1.f - expf(-sdt[q]);
        float wv    = T * alpha;
        cum += sdt[q];
        w[q] = wv;
        r0 = fmaf(wv, s_c[i * 3 + 0], r0);
        r1 = fmaf(wv, s_c[i * 3 + 1], r1);
        r2 = fmaf(wv, s_c[i * 3 + 2], r2);
    }

    // wi: 6 consecutive floats per lane, 8B-aligned -> 3x b64 stores, coalesced per wave.
    float2* wo = (float2*)(wi_out + (size_t)ray * NSAMP + i0);
    wo[0] = make_float2(w[0], w[1]);
    wo[1] = make_float2(w[2], w[3]);
    wo[2] = make_float2(w[4], w[5]);

    // rgb: wave reduction, lane 0 stores.
    #pragma unroll
    for (int off = 16; off > 0; off >>= 1) {
        r0 += __shfl_down(r0, (unsigned)off, 32);
        r1 += __shfl_down(r1, (unsigned)off, 32);
        r2 += __shfl_down(r2, (unsigned)off, 32);
    }
    if (lane == 0) {
        rgb_out[(size_t)ray * 3 + 0] = r0;
        rgb_out[(size_t)ray * 3 + 1] = r1;
        rgb_out[(size_t)ray * 3 + 2] = r2;
    }
}

extern "C" void kernel_launch(void* const* d_in, const int* in_sizes, int n_in,
                              void* d_out, int out_size, void* d_ws, size_t ws_size,
                              hipStream_t stream) {
    const float* t     = (const float*)d_in[0];   // [R, 193, 1]
    const float* sigma = (const float*)d_in[1];   // [R, 192, 1]
    const float* c     = (const float*)d_in[2];   // [R, 192, 3]
    const int R = in_sizes[0] / NT;               // 65536

    float* rgb = (float*)d_out;                   // [R,3] first
    float* wi  = rgb + (size_t)R * 3;             // then [R,192]

    nerf_integrate_kernel<<<R, 32, 0, stream>>>(t, sigma, c, rgb, wi);
}